// Aggregate_8985071583847
// MI455X (gfx1250) — compile-verified
//
#include <hip/hip_runtime.h>

// Segment-mean scatter:  out[s, :] = mean over edges e with seg(e)==s of edge_vec[e, :]
// (every segment s in [0,S) is populated by construction, and the reference's
//  segment_max scatter maps segment s to flat output row s exactly once).
//
// Memory-bound: ~2.6 GB moved -> ~110us at 23.3 TB/s. No matmul structure, so
// no WMMA; the gfx1250-specific paths used instead:
//   * global_load_async_to_lds_b128 (ASYNCcnt DMA), software-pipelined with
//     a partial s_wait_asynccnt (in-order completion lets "wait <= 8" mean
//     "previous chunk done") so HBM streaming overlaps the atomic drain
//   * th:TH_LOAD_NT on the 1 GB read-once stream so it does not evict the
//     L2-resident 164 MB accumulator (192 MB L2)
//   * single-instruction no-return global_atomic_add_f32 into L2

#define DIM 128

constexpr int EDGES_PER_CHUNK = 8;   // 8 rows * 512B = 4KB per LDS buffer
constexpr int WAVES_PER_BLOCK = 8;   // 256 threads (wave32)
constexpr int SCATTER_BLOCK   = WAVES_PER_BLOCK * 32;

__global__ void zero_kernel(float4* __restrict__ out4, int n_out4,
                            float4* __restrict__ cnt4, int n_cnt4) {
    int t = blockIdx.x * blockDim.x + threadIdx.x;
    float4 z = make_float4(0.f, 0.f, 0.f, 0.f);
    if (t < n_out4) out4[t] = z;
    if (t < n_cnt4) cnt4[t] = z;
}

__global__ __launch_bounds__(SCATTER_BLOCK)
void scatter_kernel(const float* __restrict__ edge_vec,
                    const int*   __restrict__ sel,
                    float*       __restrict__ out,
                    float*       __restrict__ counts,
                    int E) {
    // Double-buffered per-wave staging: 8 waves * 2 * 4KB = 64KB / block.
    __shared__ __align__(16) float lds[WAVES_PER_BLOCK][2][EDGES_PER_CHUNK * DIM];

    const int lane = threadIdx.x & 31;
    const int wib  = threadIdx.x >> 5;

    const int nWaves = gridDim.x * WAVES_PER_BLOCK;
    const int wid    = blockIdx.x * WAVES_PER_BLOCK + wib;
    const int stride = nWaves * EDGES_PER_CHUNK;

    // Issue n async row DMAs (512B each; this lane moves 16B of each row).
    auto issue = [&](int base, int n, float* buf) {
        for (int k = 0; k < n; ++k) {
            const float* gp = edge_vec + (size_t)(base + k) * DIM + lane * 4;
            unsigned laddr = (unsigned)(size_t)(buf + k * DIM + lane * 4);
            asm volatile("global_load_async_to_lds_b128 %0, %1, off th:TH_LOAD_NT"
                         :: "v"(laddr), "v"(gp) : "memory");
        }
    };

    int cur  = 0;
    int base = wid * EDGES_PER_CHUNK;
    if (base < E) issue(base, min(EDGES_PER_CHUNK, E - base), &lds[wib][0][0]);

    for (; base < E; base += stride) {
        const int n        = min(EDGES_PER_CHUNK, E - base);
        const int nextBase = base + stride;

        // Prefetch next chunk into the other buffer before waiting.
        int nNext = 0;
        if (nextBase < E) {
            nNext = min(EDGES_PER_CHUNK, E - nextBase);
            issue(nextBase, nNext, &lds[wib][cur ^ 1][0]);
        }

        // Segment ids for current chunk: lane k holds seg(base+k) (column 5).
        int segv = 0;
        if (lane < n) segv = sel[(size_t)(base + lane) * 6 + 5];

        // ASYNCcnt completes in order: with nNext==8 newer ops outstanding,
        // "<= 8" guarantees the current chunk's 8 DMAs have landed in LDS.
        if (nNext == EDGES_PER_CHUNK)
            asm volatile("s_wait_asynccnt 0x8" ::: "memory");
        else
            asm volatile("s_wait_asynccnt 0x0" ::: "memory");

        float* buf = &lds[wib][cur][0];
        for (int k = 0; k < n; ++k) {
            const int s = __builtin_amdgcn_readlane(segv, k);
            const float4 v = *(const float4*)(buf + k * DIM + lane * 4);
            const float* op = out + (size_t)s * DIM + lane * 4;
            // Four no-return f32 atomics sharing one 64-bit address (imm offsets).
            asm volatile(
                "global_atomic_add_f32 %0, %1, off\n\t"
                "global_atomic_add_f32 %0, %2, off offset:4\n\t"
                "global_atomic_add_f32 %0, %3, off offset:8\n\t"
                "global_atomic_add_f32 %0, %4, off offset:12"
                :: "v"(op), "v"(v.x), "v"(v.y), "v"(v.z), "v"(v.w) : "memory");
        }

        // One count increment per edge, issued in parallel by lanes 0..n-1.
        if (lane < n) {
            float one = 1.0f;
            float* cp = counts + segv;
            asm volatile("global_atomic_add_f32 %0, %1, off"
                         :: "v"(cp), "v"(one) : "memory");
        }

        cur ^= 1;
    }
}

__global__ void finalize_kernel(float4* __restrict__ out4,
                                const float* __restrict__ counts, int n4) {
    int t = blockIdx.x * blockDim.x + threadIdx.x;
    if (t >= n4) return;
    // float4 index t covers floats [4t, 4t+4) -> segment (4t)/128 = t>>5
    float c = fmaxf(counts[t >> 5], 1.0f);
    float4 v = out4[t];
    v.x /= c; v.y /= c; v.z /= c; v.w /= c;
    out4[t] = v;
}

extern "C" void kernel_launch(void* const* d_in, const int* in_sizes, int n_in,
                              void* d_out, int out_size, void* d_ws, size_t ws_size,
                              hipStream_t stream) {
    const float* edge_vec = (const float*)d_in[0];
    const int*   sel      = (const int*)d_in[1];
    float* out    = (float*)d_out;
    float* counts = (float*)d_ws;          // S floats = 1.28 MB scratch

    const int E = in_sizes[0] / DIM;       // 2,000,000 edges
    const int S = out_size / DIM;          // 320,000 segments (== B*N)
    const int n_out4 = out_size / 4;
    const int n_cnt4 = S / 4;

    {   // zero accumulator + counts (graph-replay safe: re-init every call)
        int blocks = (n_out4 + 255) / 256;
        zero_kernel<<<blocks, 256, 0, stream>>>((float4*)out, n_out4,
                                                (float4*)counts, n_cnt4);
    }

    scatter_kernel<<<2048, SCATTER_BLOCK, 0, stream>>>(edge_vec, sel, out, counts, E);

    {   // out[s,:] /= max(count[s], 1)
        int blocks = (n_out4 + 255) / 256;
        finalize_kernel<<<blocks, 256, 0, stream>>>((float4*)out, counts, n_out4);
    }
}